// AbsorbingGraphKernel_2946347565842
// MI455X (gfx1250) — compile-verified
//
#include <hip/hip_runtime.h>
#include <hip/hip_bf16.h>
#include <stdint.h>

// ---------------------------------------------------------------------------
// AbsorbingGraphKernel for MI455X (gfx1250, wave32).
// Memory-bound problem: avoid streaming logits_e/g_e (402MB) by gathering them
// only at the <=65536 selected edges. Exact per-batch top-k via 3-level radix
// select (11/11/10 bits) on monotone float keys; 32MiB key array stays L2
// resident (192MB L2). One-hot outputs built in LDS tiles and streamed out via
// the Tensor Data Mover (TENSOR_STORE_FROM_LDS), with NT-store fallback.
// ---------------------------------------------------------------------------

#define BATCH   32
#define NNODE   512
#define NCLS    6
#define MASKC   (NCLS - 1)
#define NNSQ    (NNODE * NNODE)      // 262144 edges per batch

#define PBLK    256                   // threads for scan kernels
#define PCHUNK  4096                  // elements per block
#define PITER   (PCHUNK / PBLK)       // 16
#define NCHUNK  (NNSQ / PCHUNK)       // 64

#define WROWS   4                     // rows per writer block
#define WLEN    (WROWS * NNODE * NCLS) // 12288 floats = 48KB LDS tile
#define NLEN    (NNODE * NCLS)         // 3072 floats = 12KB node tile
#define XOFF    ((size_t)BATCH * NNODE * NCLS) // Enew offset in d_out (floats)

#if defined(__has_builtin)
#  if __has_builtin(__builtin_amdgcn_tensor_store_from_lds) && \
      __has_builtin(__builtin_amdgcn_s_wait_tensorcnt)
#    define HAVE_TDM 1
#  endif
#endif
#ifndef HAVE_TDM
#  define HAVE_TDM 0
#endif

typedef unsigned int tdm_v4u __attribute__((ext_vector_type(4)));
typedef int          tdm_v8i __attribute__((ext_vector_type(8)));
typedef int          tdm_v4i __attribute__((ext_vector_type(4)));
typedef float        v4f     __attribute__((ext_vector_type(4)));

// Monotone (order-preserving) uint32 key for a float: bigger float -> bigger key.
__device__ __forceinline__ uint32_t mono_key(float f) {
    uint32_t u = __float_as_uint(f);
    uint32_t m = (u & 0x80000000u) ? 0xFFFFFFFFu : 0x80000000u;
    return u ^ m;
}

#if HAVE_TDM
// 1-D TDM store: L contiguous dwords LDS -> global. Call from a single wave
// (wave-uniform arguments); waits for the DMA before returning.
__device__ __forceinline__ void tdm_store_1d(const float* gdst,
                                             const float* ldssrc,
                                             uint32_t L) {
    const unsigned long long ga = (unsigned long long)gdst;
    const uint32_t lds = (uint32_t)(size_t)(const void*)ldssrc; // LDS byte offset
    // D# group 0: count=1 | lds_addr | global_addr[56:0] | type=2
    tdm_v4u g0 = { 1u,
                   lds,
                   (uint32_t)ga,
                   (uint32_t)((ga >> 32) & 0x01FFFFFFu) | 0x80000000u };
    // D# group 1: data_size=4B, tensor_dim0=L, tensor_dim1=1,
    //             tile_dim0=L, tile_dim1=1, dim0 stride=L.
    tdm_v8i g1 = { (int)(2u << 16),
                   (int)((L & 0xFFFFu) << 16),
                   (int)(((L >> 16) & 0xFFFFu) | (1u << 16)),
                   (int)((L & 0xFFFFu) << 16),
                   (int)1,
                   (int)L,
                   0, 0 };
    tdm_v4i gz4 = { 0, 0, 0, 0 };
    tdm_v8i gz8 = { 0, 0, 0, 0, 0, 0, 0, 0 };
    __builtin_amdgcn_tensor_store_from_lds(g0, g1, gz4, gz4, gz8, 0);
    __builtin_amdgcn_s_wait_tensorcnt(0);
}
#endif

// ---------------------------------------------------------------------------
// Nodes: one block per batch. Stable descending rank over 512 values in LDS,
// argmax over 6 classes at selected nodes, one-hot tile -> TDM store.
// ---------------------------------------------------------------------------
__global__ void k_nodes(const float* __restrict__ logits_x,
                        const float* __restrict__ g_node,
                        const float* __restrict__ g_x,
                        const int*   __restrict__ Xclass,
                        const int*   __restrict__ step_nodes,
                        float*       __restrict__ out) {
    const int b = blockIdx.x, i = threadIdx.x;
    __shared__ float lv[NNODE];
    __shared__ __align__(16) float tile[NLEN];
    __shared__ int   scnt;
    if (i == 0) scnt = 0;
    __syncthreads();
    const int  xc = Xclass[b * NNODE + i];
    const bool ab = (xc == MASKC);
    const float li = ab ? g_node[b * NNODE + i] : -1e9f;
    lv[i] = li;
    if (ab) atomicAdd(&scnt, 1);
    __syncthreads();
    const int k = min(scnt, step_nodes[0]);
    int rank = 0;
    for (int j = 0; j < NNODE; ++j) {
        float v = lv[j];
        rank += (v > li) || ((v == li) && (j < i));   // stable argsort rank
    }
    const bool sel = rank < k;
    int cls;
    if (sel) {
        const size_t p = ((size_t)b * NNODE + i) * NCLS;
        float best = logits_x[p] + g_x[p];
        cls = 0;
        #pragma unroll
        for (int d = 1; d < NCLS; ++d) {
            float v = logits_x[p + d] + g_x[p + d];
            if (v > best) { best = v; cls = d; }
        }
    } else {
        cls = xc;
    }
    #pragma unroll
    for (int d = 0; d < NCLS; ++d)
        tile[i * NCLS + d] = (d == cls) ? 1.0f : 0.0f;
    __syncthreads();
    const size_t gbase = (size_t)b * NLEN;
#if HAVE_TDM
    if (threadIdx.x < 32) tdm_store_1d(out + gbase, tile, NLEN);
#else
    const v4f* t4 = (const v4f*)tile;
    v4f* o4 = (v4f*)(out + gbase);
    for (int q = threadIdx.x; q < NLEN / 4; q += blockDim.x)
        __builtin_nontemporal_store(t4[q], &o4[q]);
#endif
}

// ---------------------------------------------------------------------------
// Edge pass 1: build keys (0 = not absorbed), level-1 histogram (top 11 bits),
// absorbed count. Streams Eclass + g_edge once with NT loads.
// ---------------------------------------------------------------------------
__global__ void k_pass1(const int*   __restrict__ Eclass,
                        const float* __restrict__ g_edge,
                        uint32_t*    __restrict__ key,
                        uint32_t*    __restrict__ hist1,
                        uint32_t*    __restrict__ cnt) {
    const int b = blockIdx.x;
    __shared__ uint32_t h[2048];
    __shared__ uint32_t sc;
    for (int x = threadIdx.x; x < 2048; x += PBLK) h[x] = 0;
    if (threadIdx.x == 0) sc = 0;
    __syncthreads();
    uint32_t c_local = 0;
    #pragma unroll
    for (int it = 0; it < PITER; ++it) {
        const int e = blockIdx.y * PCHUNK + it * PBLK + threadIdx.x;
        const size_t idx = (size_t)b * NNSQ + e;
        const int   ecls = __builtin_nontemporal_load(&Eclass[idx]);
        const float g    = __builtin_nontemporal_load(&g_edge[idx]);
        const int r = e >> 9, c = e & (NNODE - 1);
        const bool ab = (ecls == MASKC) && (c > r);
        uint32_t kk = mono_key(g);
        kk = ab ? (kk ? kk : 1u) : 0u;      // keep 0 reserved for "not absorbed"
        key[idx] = kk;
        if (ab) { atomicAdd(&h[kk >> 21], 1u); ++c_local; }
    }
    __syncthreads();
    for (int x = threadIdx.x; x < 2048; x += PBLK) {
        uint32_t v = h[x];
        if (v) atomicAdd(&hist1[(size_t)b * 2048 + x], v);
    }
    atomicAdd(&sc, c_local);
    __syncthreads();
    if (threadIdx.x == 0 && sc) atomicAdd(&cnt[b], sc);
}

// ---------------------------------------------------------------------------
// Pick the bucket holding the k-th largest key (descending scan). One block
// per batch. `first` also initializes k from cnt/step; `final_` writes the
// exact threshold key.
// ---------------------------------------------------------------------------
__global__ void k_select(const uint32_t* __restrict__ hist,
                         uint32_t* __restrict__ krem,
                         uint32_t* __restrict__ prefix,
                         uint32_t* __restrict__ done,
                         uint32_t* __restrict__ thresh,
                         const uint32_t* __restrict__ cnt,
                         const int* __restrict__ step_edges,
                         int nbuckets, int bits, int first, int final_) {
    const int b = blockIdx.x;
    __shared__ uint32_t h[2048];
    __shared__ uint32_t sdone;
    if (threadIdx.x == 0) {
        if (first) {
            const uint32_t c = cnt[b];
            const int se = step_edges[0];
            const uint32_t k = min(c, (uint32_t)(se > 0 ? se : 0));
            if (k == 0)      { done[b] = 1; thresh[b] = 0xFFFFFFFFu; }   // select none
            else if (c <= k) { done[b] = 1; thresh[b] = 1u; }            // select all absorbed
            else             { done[b] = 0; krem[b] = k; prefix[b] = 0; }
        }
        sdone = done[b];
    }
    __syncthreads();
    if (sdone) return;
    for (int x = threadIdx.x; x < nbuckets; x += blockDim.x)
        h[x] = hist[(size_t)b * nbuckets + x];
    __syncthreads();
    if (threadIdx.x == 0) {
        const uint32_t kr = krem[b];
        const uint32_t p  = prefix[b];
        uint32_t cum = 0;
        for (int x = nbuckets - 1; x >= 0; --x) {
            const uint32_t hx = h[x];
            cum += hx;
            if (cum >= kr) {
                const uint32_t np = (p << bits) | (uint32_t)x;
                if (final_) thresh[b] = np;               // exact k-th largest key
                else { prefix[b] = np; krem[b] = kr - (cum - hx); }
                break;
            }
        }
    }
}

// ---------------------------------------------------------------------------
// Refinement histogram over the key array (L2 resident) for elements whose
// high bits match the selected prefix.
// ---------------------------------------------------------------------------
__global__ void k_refine(const uint32_t* __restrict__ key,
                         const uint32_t* __restrict__ prefix,
                         const uint32_t* __restrict__ done,
                         uint32_t* __restrict__ histo,
                         int matchShift, int bucketShift, int bucketMask, int nbuckets) {
    const int b = blockIdx.x;
    if (done[b]) return;                       // uniform per block
    __shared__ uint32_t h[2048];
    for (int x = threadIdx.x; x < nbuckets; x += PBLK) h[x] = 0;
    __syncthreads();
    const uint32_t p = prefix[b];
    #pragma unroll
    for (int it = 0; it < PITER; ++it) {
        const int e = blockIdx.y * PCHUNK + it * PBLK + threadIdx.x;
        const size_t idx = (size_t)b * NNSQ + e;
        if (it + 2 < PITER)
            __builtin_prefetch(&key[idx + 2 * PBLK], 0, 1);  // global_prefetch_b8
        const uint32_t kk = key[idx];
        if (kk && ((kk >> matchShift) == p))
            atomicAdd(&h[(kk >> bucketShift) & bucketMask], 1u);
    }
    __syncthreads();
    for (int x = threadIdx.x; x < nbuckets; x += PBLK) {
        uint32_t v = h[x];
        if (v) atomicAdd(&histo[(size_t)b * nbuckets + x], v);
    }
}

// ---------------------------------------------------------------------------
// Classify: for each upper-tri edge, selected iff key >= threshold. Gather
// logits_e/g_e (6 floats each) ONLY at selected edges; otherwise keep Eclass.
// Write symmetrized class bytes (+ zero diagonal).
// ---------------------------------------------------------------------------
__global__ void k_classify(const int*   __restrict__ Eclass,
                           const float* __restrict__ logits_e,
                           const float* __restrict__ g_e,
                           const uint32_t* __restrict__ key,
                           const uint32_t* __restrict__ thresh,
                           uint8_t* __restrict__ ec) {
    const int b = blockIdx.x;
    const uint32_t th = thresh[b];
    #pragma unroll
    for (int it = 0; it < PITER; ++it) {
        const int e = blockIdx.y * PCHUNK + it * PBLK + threadIdx.x;
        const int r = e >> 9, c = e & (NNODE - 1);
        const size_t idx = (size_t)b * NNSQ + e;
        if (it + 2 < PITER)
            __builtin_prefetch(&key[idx + 2 * PBLK], 0, 1);  // global_prefetch_b8
        if (r > c) continue;                 // written by the mirror thread
        if (r == c) { ec[idx] = 0; continue; }
        const uint32_t kk = key[idx];
        uint32_t cls;
        if (kk >= th) {                      // th >= 1, so kk==0 never selected
            const size_t p = idx * NCLS;
            float best = logits_e[p] + g_e[p];
            cls = 0;
            #pragma unroll
            for (int d = 1; d < NCLS; ++d) {
                const float v = logits_e[p + d] + g_e[p + d];
                if (v > best) { best = v; cls = (uint32_t)d; }
            }
        } else {
            cls = (uint32_t)Eclass[idx];
        }
        ec[idx] = (uint8_t)cls;
        ec[(size_t)b * NNSQ + ((size_t)c << 9) + r] = (uint8_t)cls;   // mirror
    }
}

// ---------------------------------------------------------------------------
// Writer: build 4 one-hot rows (48KB) in LDS, stream to global via the
// Tensor Data Mover (one 12288-dword 1D tile per block), NT-store fallback.
// ---------------------------------------------------------------------------
__global__ void k_writer(const uint8_t* __restrict__ ec,
                         float* __restrict__ out) {
    __shared__ __align__(16) float tile[WLEN];
    const int x  = blockIdx.x;
    const int b  = x >> 7;                 // 128 blocks per batch (512 rows / 4)
    const int r0 = (x & 127) * WROWS;
    const int t  = threadIdx.x;            // 512 threads = one column each
    #pragma unroll
    for (int row = 0; row < WROWS; ++row) {
        const int r = r0 + row;
        const int cls = (int)ec[(size_t)b * NNSQ + (size_t)r * NNODE + t];
        const bool diag = (r == t);
        #pragma unroll
        for (int d = 0; d < NCLS; ++d)
            tile[row * (NNODE * NCLS) + t * NCLS + d] =
                (!diag && d == cls) ? 1.0f : 0.0f;
    }
    __syncthreads();
    const size_t gbase = XOFF + ((size_t)b * NNSQ + (size_t)r0 * NNODE) * NCLS;
#if HAVE_TDM
    if (threadIdx.x < 32) tdm_store_1d(out + gbase, tile, WLEN);
#else
    const v4f* t4 = (const v4f*)tile;
    v4f* o4 = (v4f*)(out + gbase);
    for (int q = threadIdx.x; q < WLEN / 4; q += blockDim.x)
        __builtin_nontemporal_store(t4[q], &o4[q]);
#endif
}

// ---------------------------------------------------------------------------
extern "C" void kernel_launch(void* const* d_in, const int* in_sizes, int n_in,
                              void* d_out, int out_size, void* d_ws, size_t ws_size,
                              hipStream_t stream) {
    const float* logits_x   = (const float*)d_in[0];
    const float* logits_e   = (const float*)d_in[1];
    const float* g_node     = (const float*)d_in[2];
    const float* g_x        = (const float*)d_in[3];
    const float* g_edge     = (const float*)d_in[4];
    const float* g_e        = (const float*)d_in[5];
    const int*   Xclass     = (const int*)d_in[6];
    const int*   Eclass     = (const int*)d_in[7];
    const int*   step_nodes = (const int*)d_in[8];
    const int*   step_edges = (const int*)d_in[9];
    float* out = (float*)d_out;

    // Workspace carve-up.
    uint32_t* key   = (uint32_t*)d_ws;                               // 32 MiB
    uint8_t*  ec    = (uint8_t*)(key + (size_t)BATCH * NNSQ);        // 8 MiB
    uint32_t* hist1 = (uint32_t*)(ec + (size_t)BATCH * NNSQ);
    uint32_t* hist2 = hist1 + (size_t)BATCH * 2048;
    uint32_t* hist3 = hist2 + (size_t)BATCH * 2048;
    uint32_t* cnt   = hist3 + (size_t)BATCH * 1024;
    uint32_t* krem  = cnt    + BATCH;
    uint32_t* prefix= krem   + BATCH;
    uint32_t* done  = prefix + BATCH;
    uint32_t* thresh= done   + BATCH;
    const size_t zero_bytes = ((size_t)BATCH * (2048 + 2048 + 1024) + 5 * BATCH)
                              * sizeof(uint32_t);
    (void)hipMemsetAsync(hist1, 0, zero_bytes, stream);

    // Nodes (tiny).
    k_nodes<<<BATCH, NNODE, 0, stream>>>(logits_x, g_node, g_x, Xclass,
                                         step_nodes, out);

    // Edge top-k: keys + 3-level radix select.
    dim3 sg(BATCH, NCHUNK);
    k_pass1<<<sg, PBLK, 0, stream>>>(Eclass, g_edge, key, hist1, cnt);
    k_select<<<BATCH, PBLK, 0, stream>>>(hist1, krem, prefix, done, thresh,
                                         cnt, step_edges, 2048, 11, 1, 0);
    k_refine<<<sg, PBLK, 0, stream>>>(key, prefix, done, hist2, 21, 10, 0x7FF, 2048);
    k_select<<<BATCH, PBLK, 0, stream>>>(hist2, krem, prefix, done, thresh,
                                         cnt, step_edges, 2048, 11, 0, 0);
    k_refine<<<sg, PBLK, 0, stream>>>(key, prefix, done, hist3, 10, 0, 0x3FF, 1024);
    k_select<<<BATCH, PBLK, 0, stream>>>(hist3, krem, prefix, done, thresh,
                                         cnt, step_edges, 1024, 10, 0, 1);

    // Classify (sparse gather of logits_e/g_e) + symmetrize class bytes.
    k_classify<<<sg, PBLK, 0, stream>>>(Eclass, logits_e, g_e, key, thresh, ec);

    // One-hot stream-out (201 MB) via LDS tiles + TDM / NT stores.
    k_writer<<<BATCH * NNODE / WROWS, NNODE, 0, stream>>>(ec, out);
}